// GATGAE_37426345017688
// MI455X (gfx1250) — compile-verified
//
#include <hip/hip_runtime.h>
#include <math.h>

// ---------------- problem constants (fixed by the reference) ----------------
constexpr int NN = 30000;
constexpr int EE = 480000;
constexpr int E2C = EE + NN;           // edges + self loops
constexpr int H1 = 4, D1 = 64, F1 = 256;   // layer1 heads*dim
constexpr int D2 = 32;                  // layer2 out dim (1 head)
constexpr int NPART = (EE + 255) / 256; // 1875 BN partial blocks

// ---------------- workspace layout (float offsets) ----------------
constexpr size_t OFF_EA    = 0;                                   // EE
constexpr size_t OFF_ZERO  = OFF_EA + EE;                         // zero-init block start
constexpr size_t OFF_SUMEA = OFF_ZERO;                            // NN
constexpr size_t OFF_CNT   = OFF_SUMEA + NN;                      // NN
constexpr size_t OFF_M1    = OFF_CNT + NN;                        // NN*4 (encoded uint)
constexpr size_t OFF_S1    = OFF_M1 + (size_t)NN * 4;             // NN*4
constexpr size_t OFF_OUT1  = OFF_S1 + (size_t)NN * 4;             // NN*256
constexpr size_t OFF_M2    = OFF_OUT1 + (size_t)NN * F1;          // NN (encoded uint)
constexpr size_t OFF_S2    = OFF_M2 + NN;                         // NN
constexpr size_t OFF_OUT2  = OFF_S2 + NN;                         // NN*32
constexpr size_t ZERO_CNT  = OFF_OUT2 + (size_t)NN * D2 - OFF_ZERO;
constexpr size_t OFF_LOOP  = OFF_OUT2 + (size_t)NN * D2;          // NN
constexpr size_t OFF_H1    = OFF_LOOP + NN;                       // NN*256
constexpr size_t OFF_SSRC1 = OFF_H1 + (size_t)NN * F1;            // NN*4
constexpr size_t OFF_SDST1 = OFF_SSRC1 + (size_t)NN * 4;          // NN*4
constexpr size_t OFF_EX1   = OFF_SDST1 + (size_t)NN * 4;          // E2*4
constexpr size_t OFF_ZPRE  = OFF_EX1 + (size_t)E2C * 4;           // NN*32
constexpr size_t OFF_SSRC2 = OFF_ZPRE + (size_t)NN * D2;          // NN
constexpr size_t OFF_SDST2 = OFF_SSRC2 + NN;                      // NN
constexpr size_t OFF_EX2   = OFF_SDST2 + NN;                      // E2
constexpr size_t OFF_DECH  = OFF_EX2 + (size_t)E2C;               // NN*64
constexpr size_t OFF_PART  = OFF_DECH + (size_t)NN * 64;          // 2*NPART
constexpr size_t OFF_RED   = OFF_PART + (size_t)2 * NPART;        // 16 (mu,rstd,aedot1[4],aedot2)

#define CDIV(a, b) (((a) + (b) - 1) / (b))

typedef float v2f __attribute__((ext_vector_type(2)));
typedef float v8f __attribute__((ext_vector_type(8)));

// orderable-uint encoding for float atomic max
__device__ __forceinline__ unsigned encf(float f) {
  unsigned u = __float_as_uint(f);
  return (u & 0x80000000u) ? ~u : (u | 0x80000000u);
}
__device__ __forceinline__ float decf(unsigned u) {
  return __uint_as_float((u & 0x80000000u) ? (u & 0x7FFFFFFFu) : ~u);
}

// ---------------- zero init ----------------
__global__ void k_zero(float* p, long long n) {
  long long i = (long long)blockIdx.x * blockDim.x + threadIdx.x;
  if (i < n) p[i] = 0.0f;
}

// ---------------- BN stats (deterministic tree reduction) ----------------
__global__ void k_bn_partial(const float* __restrict__ ea, float* __restrict__ part) {
  __shared__ float sh[256], sh2[256];
  int i = blockIdx.x * 256 + threadIdx.x;
  float v = 0.0f;
  if (i < EE) v = log1pf(ea[i]);
  sh[threadIdx.x] = v;
  sh2[threadIdx.x] = v * v;
  __syncthreads();
  for (int s = 128; s > 0; s >>= 1) {
    if (threadIdx.x < s) {
      sh[threadIdx.x] += sh[threadIdx.x + s];
      sh2[threadIdx.x] += sh2[threadIdx.x + s];
    }
    __syncthreads();
  }
  if (threadIdx.x == 0) {
    part[2 * blockIdx.x] = sh[0];
    part[2 * blockIdx.x + 1] = sh2[0];
  }
}

__global__ void k_bn_final(const float* __restrict__ part, float* __restrict__ red) {
  __shared__ float sh[256], sh2[256];
  float s = 0.0f, q = 0.0f;
  for (int i = threadIdx.x; i < NPART; i += 256) {
    s += part[2 * i];
    q += part[2 * i + 1];
  }
  sh[threadIdx.x] = s;
  sh2[threadIdx.x] = q;
  __syncthreads();
  for (int st = 128; st > 0; st >>= 1) {
    if (threadIdx.x < st) {
      sh[threadIdx.x] += sh[threadIdx.x + st];
      sh2[threadIdx.x] += sh2[threadIdx.x + st];
    }
    __syncthreads();
  }
  if (threadIdx.x == 0) {
    float mu = sh[0] / (float)EE;
    float var = sh2[0] / (float)EE - mu * mu;  // biased var
    red[0] = mu;
    red[1] = rsqrtf(var + 1e-5f);
  }
}

// ae_dot[h] = sum_d We[0,h*D+d]*a_e[h,d] (edge attn collapses since EDIM==1)
__global__ void k_attn_consts(const float* __restrict__ We1, const float* __restrict__ ae1,
                              const float* __restrict__ We2, const float* __restrict__ ae2,
                              float* __restrict__ red) {
  __shared__ float sh[256];
  int t = threadIdx.x;
  sh[t] = We1[t] * ae1[t];
  __syncthreads();
  if (t < 4) {
    float s = 0.0f;
    for (int d = 0; d < 64; ++d) s += sh[t * 64 + d];
    red[2 + t] = s;
  }
  __syncthreads();
  sh[t] = (t < 32) ? We2[t] * ae2[t] : 0.0f;
  __syncthreads();
  if (t == 0) {
    float s = 0.0f;
    for (int d = 0; d < 32; ++d) s += sh[d];
    red[6] = s;
  }
}

// ---------------- edge normalization + per-dst sums for self-loop fill ----------------
__global__ void k_norm_edges(const float* __restrict__ eattr, const float* __restrict__ gamma,
                             const float* __restrict__ beta, const float* __restrict__ red,
                             const int* __restrict__ dstA, float* __restrict__ ea_norm,
                             float* __restrict__ sum_ea, float* __restrict__ cnt) {
  int i = blockIdx.x * blockDim.x + threadIdx.x;
  if (i >= EE) return;
  float v = (log1pf(eattr[i]) - red[0]) * red[1] * gamma[0] + beta[0];
  ea_norm[i] = v;
  int d = dstA[i];
  atomicAdd(&cnt[d], 1.0f);
  atomicAdd(&sum_ea[d], v);
}

__global__ void k_loop_attr(const float* __restrict__ sum_ea, const float* __restrict__ cnt,
                            float* __restrict__ loop) {
  int n = blockIdx.x * blockDim.x + threadIdx.x;
  if (n < NN) loop[n] = sum_ea[n] / fmaxf(cnt[n], 1.0f);
}

// ---------------- WMMA f32 GEMM: C[M,N] = A[M,K] @ B[K,N], row-major ----------------
// one wave per 16x16 tile; act: 0=none, 1=ELU
__global__ void k_wmma_gemm_f32(const float* __restrict__ A, const float* __restrict__ B,
                                float* __restrict__ C, int M, int N, int K,
                                const float* __restrict__ bias, int act) {
  int wave = blockIdx.x * (blockDim.x >> 5) + (threadIdx.x >> 5);
  int lane = threadIdx.x & 31;
  int tilesN = N >> 4;
  int mt = wave / tilesN;
  int nt = wave - mt * tilesN;
  if (mt * 16 >= M) return;  // uniform across the wave

  int half = lane >> 4;          // 0: lanes 0-15, 1: lanes 16-31
  int l15 = lane & 15;
  // A frag: row = mt*16 + (lane&15); K pair base = 2*half
  const float* Ap = A + (size_t)(mt * 16 + l15) * K + 2 * half;
  // B frag: col = nt*16 + (lane&15); K pair base = 2*half
  const float* Bp = B + (size_t)(2 * half) * N + nt * 16 + l15;

  v8f acc = {};
  for (int k = 0; k < K; k += 4) {
    v2f a, b;
    a.x = Ap[k];
    a.y = Ap[k + 1];
    b.x = Bp[(size_t)k * N];
    b.y = Bp[(size_t)(k + 1) * N];
    acc = __builtin_amdgcn_wmma_f32_16x16x4_f32(false, a, false, b, (short)0, acc,
                                                false, false);
  }

  // C/D layout: VGPR i -> row mt*16 + i + 8*half, col nt*16 + (lane&15)
  int r0 = mt * 16 + 8 * half;
  int c = nt * 16 + l15;
  float bv = bias ? bias[c] : 0.0f;
#pragma unroll
  for (int i = 0; i < 8; ++i) {
    float v = acc[i] + bv;
    if (act == 1) v = (v > 0.0f) ? v : (expf(v) - 1.0f);
    C[(size_t)(r0 + i) * N + c] = v;
  }
}

// ---------------- per-node attention dot products ----------------
__global__ void k_node_dots(const float* __restrict__ hmat, const float* __restrict__ a_s,
                            const float* __restrict__ a_d, float* __restrict__ ssrc,
                            float* __restrict__ sdst, int H, int D) {
  int t = blockIdx.x * blockDim.x + threadIdx.x;
  if (t >= NN * H) return;
  int n = t / H, h = t - n * H;
  const float* hp = hmat + (size_t)n * H * D + (size_t)h * D;
  float s1 = 0.0f, s2 = 0.0f;
  for (int d = 0; d < D; ++d) {
    float v = hp[d];
    s1 += v * a_s[h * D + d];
    s2 += v * a_d[h * D + d];
  }
  ssrc[t] = s1;
  sdst[t] = s2;
}

// ---------------- softmax pass 1: logits + leakyrelu + segment max ----------------
__global__ void k_edge_pass1(const int* __restrict__ srcA, const int* __restrict__ dstA,
                             const float* __restrict__ ea_norm, const float* __restrict__ loop,
                             const float* __restrict__ ssrc, const float* __restrict__ sdst,
                             const float* __restrict__ aedot, float* __restrict__ logits,
                             unsigned* __restrict__ mbuf, int H) {
  int t = blockIdx.x * blockDim.x + threadIdx.x;
  if (t >= E2C * H) return;
  int e = t / H, h = t - e * H;
  int s_, d_;
  float eav;
  if (e < EE) {
    s_ = srcA[e];
    d_ = dstA[e];
    eav = ea_norm[e];
  } else {
    s_ = d_ = e - EE;
    eav = loop[s_];
  }
  float lg = ssrc[s_ * H + h] + sdst[d_ * H + h] + eav * aedot[h];
  lg = (lg > 0.0f) ? lg : 0.2f * lg;  // leaky relu
  logits[t] = lg;
  atomicMax(mbuf + (size_t)d_ * H + h, encf(lg));
}

// ---------------- softmax pass 2: exp + segment sum (in-place logits -> ex) ----------------
__global__ void k_edge_pass2(const int* __restrict__ dstA, float* __restrict__ logits,
                             const unsigned* __restrict__ mbuf, float* __restrict__ sbuf, int H) {
  int t = blockIdx.x * blockDim.x + threadIdx.x;
  if (t >= E2C * H) return;
  int e = t / H, h = t - e * H;
  int d_ = (e < EE) ? dstA[e] : e - EE;
  float ex = expf(logits[t] - decf(mbuf[(size_t)d_ * H + h]));
  logits[t] = ex;
  atomicAdd(sbuf + (size_t)d_ * H + h, ex);
}

// ---------------- softmax pass 3: weighted scatter aggregation ----------------
// one block per edge, blockDim == H*D
__global__ void k_edge_aggregate(const int* __restrict__ srcA, const int* __restrict__ dstA,
                                 const float* __restrict__ ex, const float* __restrict__ sbuf,
                                 const float* __restrict__ hmat, float* __restrict__ outm,
                                 int H, int D) {
  int e = blockIdx.x;
  int t = threadIdx.x;
  int h = t / D;
  int s_, d_;
  if (e < EE) {
    s_ = srcA[e];
    d_ = dstA[e];
  } else {
    s_ = d_ = e - EE;
  }
  int F = H * D;
  float alpha = ex[(size_t)e * H + h] / (sbuf[(size_t)d_ * H + h] + 1e-16f);
  atomicAdd(&outm[(size_t)d_ * F + t], alpha * hmat[(size_t)s_ * F + t]);
}

// ---------------- epilogues ----------------
__global__ void k_bias_elu_inplace(float* __restrict__ p, const float* __restrict__ bias,
                                   long long n, int F) {
  long long i = (long long)blockIdx.x * blockDim.x + threadIdx.x;
  if (i >= n) return;
  float v = p[i] + bias[i % F];
  p[i] = (v > 0.0f) ? v : (expf(v) - 1.0f);
}

__global__ void k_bias_copy(const float* __restrict__ src, const float* __restrict__ bias,
                            float* __restrict__ dst, long long n, int F) {
  long long i = (long long)blockIdx.x * blockDim.x + threadIdx.x;
  if (i >= n) return;
  dst[i] = src[i] + bias[i % F];
}

// ======================= launch =======================
extern "C" void kernel_launch(void* const* d_in, const int* in_sizes, int n_in,
                              void* d_out, int out_size, void* d_ws, size_t ws_size,
                              hipStream_t stream) {
  const float* x = (const float*)d_in[0];
  const float* eattr = (const float*)d_in[1];
  const float* gamma = (const float*)d_in[2];
  const float* beta = (const float*)d_in[3];
  const float* W1 = (const float*)d_in[4];
  const float* We1 = (const float*)d_in[5];
  const float* as1 = (const float*)d_in[6];
  const float* ad1 = (const float*)d_in[7];
  const float* ae1 = (const float*)d_in[8];
  const float* b1 = (const float*)d_in[9];
  const float* W2 = (const float*)d_in[10];
  const float* We2 = (const float*)d_in[11];
  const float* as2 = (const float*)d_in[12];
  const float* ad2 = (const float*)d_in[13];
  const float* ae2 = (const float*)d_in[14];
  const float* b2 = (const float*)d_in[15];
  const float* dW1 = (const float*)d_in[16];
  const float* db1 = (const float*)d_in[17];
  const float* dW2 = (const float*)d_in[18];
  const float* db2 = (const float*)d_in[19];
  const int* ei = (const int*)d_in[20];
  const int* srcA = ei;
  const int* dstA = ei + EE;

  float* ws = (float*)d_ws;
  float* ea_norm = ws + OFF_EA;
  float* sum_ea = ws + OFF_SUMEA;
  float* cnt = ws + OFF_CNT;
  unsigned* m1 = (unsigned*)(ws + OFF_M1);
  float* s1 = ws + OFF_S1;
  float* out1 = ws + OFF_OUT1;
  unsigned* m2 = (unsigned*)(ws + OFF_M2);
  float* s2 = ws + OFF_S2;
  float* out2 = ws + OFF_OUT2;
  float* loop = ws + OFF_LOOP;
  float* h1 = ws + OFF_H1;
  float* ssrc1 = ws + OFF_SSRC1;
  float* sdst1 = ws + OFF_SDST1;
  float* ex1 = ws + OFF_EX1;
  float* zpre = ws + OFF_ZPRE;
  float* ssrc2 = ws + OFF_SSRC2;
  float* sdst2 = ws + OFF_SDST2;
  float* ex2 = ws + OFF_EX2;
  float* dech = ws + OFF_DECH;
  float* part = ws + OFF_PART;
  float* red = ws + OFF_RED;

  float* z_out = (float*)d_out;              // [NN, 32]
  float* xhat_out = z_out + (size_t)NN * 32; // [NN, 128]

  // 0) zero accumulator block (sum_ea,cnt,m1,s1,out1,m2,s2,out2); m encoding: 0 == -inf floor
  k_zero<<<CDIV((long long)ZERO_CNT, 256), 256, 0, stream>>>(ws + OFF_ZERO, (long long)ZERO_CNT);

  // 1) BN stats + attention constants
  k_bn_partial<<<NPART, 256, 0, stream>>>(eattr, part);
  k_bn_final<<<1, 256, 0, stream>>>(part, red);
  k_attn_consts<<<1, 256, 0, stream>>>(We1, ae1, We2, ae2, red);

  // 2) normalize edge attrs, per-dst count/sum, self-loop fill
  k_norm_edges<<<CDIV(EE, 256), 256, 0, stream>>>(eattr, gamma, beta, red, dstA, ea_norm,
                                                  sum_ea, cnt);
  k_loop_attr<<<CDIV(NN, 256), 256, 0, stream>>>(sum_ea, cnt, loop);

  // 3) layer-1 GEMM: h1 = x @ W1  (30000x128 @ 128x256) -- WMMA f32
  {
    int waves = (NN / 16) * (F1 / 16);
    k_wmma_gemm_f32<<<CDIV(waves, 8), 256, 0, stream>>>(x, W1, h1, NN, F1, 128, nullptr, 0);
  }

  // 4) layer-1 attention
  k_node_dots<<<CDIV(NN * H1, 256), 256, 0, stream>>>(h1, as1, ad1, ssrc1, sdst1, H1, D1);
  k_edge_pass1<<<CDIV(E2C * H1, 256), 256, 0, stream>>>(srcA, dstA, ea_norm, loop, ssrc1,
                                                        sdst1, red + 2, ex1, m1, H1);
  k_edge_pass2<<<CDIV(E2C * H1, 256), 256, 0, stream>>>(dstA, ex1, m1, s1, H1);
  k_edge_aggregate<<<E2C, F1, 0, stream>>>(srcA, dstA, ex1, s1, h1, out1, H1, D1);
  k_bias_elu_inplace<<<CDIV((long long)NN * F1, 256), 256, 0, stream>>>(out1, b1,
                                                                        (long long)NN * F1, F1);

  // 5) layer-2 GEMM: zpre = h1act @ W2 (30000x256 @ 256x32) -- WMMA f32
  {
    int waves = (NN / 16) * (D2 / 16);
    k_wmma_gemm_f32<<<CDIV(waves, 8), 256, 0, stream>>>(out1, W2, zpre, NN, D2, F1, nullptr, 0);
  }

  // 6) layer-2 attention (1 head)
  k_node_dots<<<CDIV(NN, 256), 256, 0, stream>>>(zpre, as2, ad2, ssrc2, sdst2, 1, D2);
  k_edge_pass1<<<CDIV(E2C, 256), 256, 0, stream>>>(srcA, dstA, ea_norm, loop, ssrc2, sdst2,
                                                   red + 6, ex2, m2, 1);
  k_edge_pass2<<<CDIV(E2C, 256), 256, 0, stream>>>(dstA, ex2, m2, s2, 1);
  k_edge_aggregate<<<E2C, D2, 0, stream>>>(srcA, dstA, ex2, s2, zpre, out2, 1, D2);
  k_bias_copy<<<CDIV((long long)NN * D2, 256), 256, 0, stream>>>(out2, b2, z_out,
                                                                 (long long)NN * D2, D2);

  // 7) decoder: x_hat = elu(z @ dec_W1 + b) @ dec_W2 + b  -- WMMA f32 x2
  {
    int waves = (NN / 16) * (64 / 16);
    k_wmma_gemm_f32<<<CDIV(waves, 8), 256, 0, stream>>>(z_out, dW1, dech, NN, 64, 32, db1, 1);
  }
  {
    int waves = (NN / 16) * (128 / 16);
    k_wmma_gemm_f32<<<CDIV(waves, 8), 256, 0, stream>>>(dech, dW2, xhat_out, NN, 128, 64, db2, 0);
  }
}